// DeformableConv_76922864271988
// MI455X (gfx1250) — compile-verified
//
#include <hip/hip_runtime.h>

typedef __attribute__((ext_vector_type(2))) float v2f;
typedef __attribute__((ext_vector_type(8))) float v8f;

#define KH    3
#define KW    3
#define KK    9
#define C_IN  256
#define C_OUT 256
#define H_IN  64
#define W_IN  64
#define HW    (H_IN * W_IN)
#define KTOT  (C_IN * KK)     // 2304 = GEMM K
#define CC    32              // channels per LDS chunk
#define KSPAN (CC * KK)       // 288 K-rows per chunk
#define NPIX  64              // one output row per block
#define NMETA (KK * NPIX)     // 576 (k,pixel) entries

__global__ __launch_bounds__(256)
void dcn_fused_wmma(const float* __restrict__ input,
                    const float* __restrict__ offset,
                    const float* __restrict__ weight,
                    float* __restrict__ out)
{
    __shared__ float s_col[KSPAN * NPIX];   // 73728 B, k-major [Krow][pix]
    __shared__ int   s_idx[4 * NMETA];      //  9216 B, clamped corner indices
    __shared__ float s_wgt[4 * NMETA];      //  9216 B, masked corner weights

    const int tid  = threadIdx.x;
    const int bimg = blockIdx.x >> 6;   // batch
    const int hrow = blockIdx.x & 63;   // output row

    // ---------- Phase 0: per-(k,pixel) bilinear metadata (reused for all C) ----------
    for (int e = tid; e < NMETA; e += 256) {
        const int k  = e >> 6;          // 0..8
        const int px = e & 63;          // output column
        const int ky = k / KW, kx = k % KW;
        const float offy = offset[(((size_t)bimg * (2 * KK) + (2 * k + 0)) * H_IN + hrow) * W_IN + px];
        const float offx = offset[(((size_t)bimg * (2 * KK) + (2 * k + 1)) * H_IN + hrow) * W_IN + px];
        const float y = (float)(ky + hrow - 1) + offy;   // dil=1, stride=1, pad=1
        const float x = (float)(kx + px   - 1) + offx;
        const float y0f = floorf(y), x0f = floorf(x);
        const float wy1 = y - y0f, wx1 = x - x0f;
        const float wy0 = 1.0f - wy1, wx0 = 1.0f - wx1;
        const int y0 = (int)y0f, x0 = (int)x0f;
#pragma unroll
        for (int c4 = 0; c4 < 4; ++c4) {
            const int dy = c4 >> 1, dx = c4 & 1;
            const int yi = y0 + dy, xi = x0 + dx;
            const bool valid = (yi >= 0) && (yi < H_IN) && (xi >= 0) && (xi < W_IN);
            const int yc = min(max(yi, 0), H_IN - 1);
            const int xc = min(max(xi, 0), W_IN - 1);
            const float w = (dy ? wy1 : wy0) * (dx ? wx1 : wx0);
            s_idx[c4 * NMETA + e] = yc * W_IN + xc;
            s_wgt[c4 * NMETA + e] = valid ? w : 0.0f;
        }
    }
    __syncthreads();

    // Wave tiling: 4 M-waves (64 out-ch each) x 2 N-waves (32 px each)
    const int lane = tid & 31;
    const int wave = tid >> 5;
    const int mw = wave & 3;
    const int nw = wave >> 2;
    const int hs = lane >> 4;    // K half-select per ISA 16x4 fp32 A layout
    const int ln = lane & 15;

    v8f acc[4][2] = {};          // 16x16 f32 C tiles, 8 VGPRs each

    for (int cb = 0; cb < C_IN; cb += CC) {
        // ------- Phase 1: deformable im2col chunk into LDS -------
        const float* plane0 = input + ((size_t)bimg * C_IN + cb) * HW;
        for (int e = tid; e < NMETA; e += 256) {
            const int k  = e >> 6;
            const int px = e & 63;
            const int i00 = s_idx[0 * NMETA + e], i01 = s_idx[1 * NMETA + e];
            const int i10 = s_idx[2 * NMETA + e], i11 = s_idx[3 * NMETA + e];
            const float w00 = s_wgt[0 * NMETA + e], w01 = s_wgt[1 * NMETA + e];
            const float w10 = s_wgt[2 * NMETA + e], w11 = s_wgt[3 * NMETA + e];
            float* colw = &s_col[k * NPIX + px];    // row (c*9+k), stride 576 floats
            const float* p = plane0;
#pragma unroll 4
            for (int c = 0; c < CC; ++c) {
                const float v = fmaf(w00, p[i00],
                                fmaf(w01, p[i01],
                                fmaf(w10, p[i10], w11 * p[i11])));
                colw[c * (KK * NPIX)] = v;
                p += HW;
            }
        }
        __syncthreads();

        // ------- Phase 2: WMMA f32 16x16x4 GEMM over this K chunk -------
        const float* wchunk = weight + cb * KK;     // K-dim column offset
#pragma unroll 2
        for (int kb = 0; kb < KSPAN / 4; ++kb) {
            const int krow = kb * 4 + hs * 2;       // this half-wave's K pair
            v2f bfr[2];
#pragma unroll
            for (int nt = 0; nt < 2; ++nt) {
                const int ncol = nw * 32 + nt * 16 + ln;
                bfr[nt].x = s_col[(krow + 0) * NPIX + ncol];
                bfr[nt].y = s_col[(krow + 1) * NPIX + ncol];
            }
            v2f afr[4];
#pragma unroll
            for (int mt = 0; mt < 4; ++mt) {
                const int m = mw * 64 + mt * 16 + ln;
                afr[mt] = *(const v2f*)(wchunk + (size_t)m * KTOT + krow);
            }
#pragma unroll
            for (int mt = 0; mt < 4; ++mt)
#pragma unroll
                for (int nt = 0; nt < 2; ++nt)
                    acc[mt][nt] = __builtin_amdgcn_wmma_f32_16x16x4_f32(
                        false, afr[mt], false, bfr[nt],
                        (short)0, acc[mt][nt], false, false);
        }
        __syncthreads();
    }

    // ------- Phase 3: store C tiles (ISA 16x16 f32 C/D layout) -------
#pragma unroll
    for (int mt = 0; mt < 4; ++mt) {
#pragma unroll
        for (int nt = 0; nt < 2; ++nt) {
            const int n = nw * 32 + nt * 16 + ln;
#pragma unroll
            for (int r = 0; r < 8; ++r) {
                const int m = mw * 64 + mt * 16 + hs * 8 + r;
                out[(((size_t)bimg * C_OUT + m) * H_IN + hrow) * W_IN + n] = acc[mt][nt][r];
            }
        }
    }
}

extern "C" void kernel_launch(void* const* d_in, const int* in_sizes, int n_in,
                              void* d_out, int out_size, void* d_ws, size_t ws_size,
                              hipStream_t stream)
{
    const float* input  = (const float*)d_in[0];   // (8,256,64,64)
    const float* offset = (const float*)d_in[1];   // (8,18,64,64)
    const float* weight = (const float*)d_in[2];   // (256,256,3,3)
    float* out = (float*)d_out;                    // (8,256,64,64)

    const int nblocks = 8 * H_IN;                  // one block per (batch, row)
    dcn_fused_wmma<<<nblocks, 256, 0, stream>>>(input, offset, weight, out);
}